// AveragedIWE_1709396984512
// MI455X (gfx1250) — compile-verified
//
#include <hip/hip_runtime.h>
#include <stdint.h>

#define H_DIM 128
#define W_DIM 160
#define HW_DIM (H_DIM * W_DIM)      // 20480
#define FLOW_SCALE 160.0f           // max(H, W)
#define WAVES_PER_BLOCK 8           // 256 threads = 8 wave32s

// ---------------------------------------------------------------------------
// Kernel 1: zero the workspace (acc floats, contrib u32, hash table u32).
// All region sizes are multiples of 4 DWORDs -> vectorized uint4 stores.
// ---------------------------------------------------------------------------
__global__ void zero_ws_kernel(uint4* __restrict__ ws, size_t nquads) {
    size_t i = (size_t)blockIdx.x * blockDim.x + threadIdx.x;
    size_t stride = (size_t)gridDim.x * blockDim.x;
    uint4 z = make_uint4(0u, 0u, 0u, 0u);
    for (; i < nquads; i += stride) ws[i] = z;
}

// ---------------------------------------------------------------------------
// Kernel 2: per-event warp + scatter.
// Each wave32 stages its 32-event (512B) tile into LDS via the CDNA5 async
// Global->LDS path (ASYNCcnt), then processes one event per lane:
//   - gather flow at source pixel (L2-resident, 655KB total)
//   - warp, round-half-even (rintf -> v_rndne_f32, matches jnp.round)
//   - atomicAdd 1.0f into per-(batch,pol) histogram
//   - dedup (pol,src,dst) pair in a per-batch hash set; first insert bumps
//     the contrib count for that (pol,dst) bin.
// ---------------------------------------------------------------------------
__global__ __launch_bounds__(WAVES_PER_BLOCK * 32)
void iwe_scatter_kernel(const float* __restrict__ flow,      // (B,2,H,W)
                        const float* __restrict__ events,    // (B,N,4)
                        float* __restrict__ acc,             // (B,2,HW)
                        uint32_t* __restrict__ contrib,      // (B,2,HW)
                        uint32_t* __restrict__ table,        // (B,cap)
                        int nPerBatch, int nTotal,
                        uint32_t cap_mask, uint32_t hash_shift)
{
    __shared__ float4 buf[WAVES_PER_BLOCK][32];

    const int lane = threadIdx.x & 31;
    const int wv   = threadIdx.x >> 5;
    const int tile = blockIdx.x * WAVES_PER_BLOCK + wv;
    const int i0   = tile * 32;
    if (i0 >= nTotal) return;                 // wave-uniform exit

    const int i      = i0 + lane;
    const int iclamp = (i < nTotal) ? i : (nTotal - 1);

    // --- CDNA5 async Global -> LDS stage of this wave's 32 events (512B) ---
    uint32_t lds_addr = (uint32_t)(uintptr_t)(&buf[wv][lane]);   // low 32 bits of
                                                                 // generic ptr == LDS offset
    uint64_t gaddr = (uint64_t)(uintptr_t)((const float4*)events + iclamp);
    asm volatile("global_load_async_to_lds_b128 %0, %1, off"
                 :: "v"(lds_addr), "v"(gaddr) : "memory");
    asm volatile("s_wait_asynccnt 0x0" ::: "memory");

    float4 ev = buf[wv][lane];                // {ts, y, x, p}
    if (i >= nTotal) return;

    const float ts = ev.x, yf = ev.y, xf = ev.z, p = ev.w;

    const int b = i / nPerBatch;
    const float* __restrict__ fb = flow + (size_t)b * (2 * HW_DIM);

    const int s = (int)yf * W_DIM + (int)xf;          // source flat index
    const float dt = (1.0f - ts) * FLOW_SCALE;
    const float wy = yf + dt * fb[HW_DIM + s];        // channel 1 = fy
    const float wx = xf + dt * fb[s];                 // channel 0 = fx
    const float ry = rintf(wy);                       // round-half-even
    const float rx = rintf(wx);

    const bool valid = (ry >= 0.0f) && (ry < (float)H_DIM) &&
                       (rx >= 0.0f) && (rx < (float)W_DIM);
    if (!valid) return;   // invalid events contribute nothing (pol_id==2 bucket
                          // is discarded and weight==0 in the reference)

    const int f  = (int)ry * W_DIM + (int)rx;         // dest flat index
    const int ch = (p > 0.0f) ? 0 : 1;                // 0 = pos, 1 = neg
    const size_t bin = ((size_t)b * 2 + (size_t)ch) * HW_DIM + (size_t)f;

    atomicAdd(&acc[bin], 1.0f);                       // exact integer-valued sum

    // --- distinct (pol, src, dst) pair dedup via per-batch hash set ---
    const uint32_t key = ((uint32_t)(ch * HW_DIM + s)) * (uint32_t)HW_DIM
                         + (uint32_t)f + 1u;          // +1 so 0 == empty
    uint32_t* __restrict__ tab = table + (size_t)b * (size_t)(cap_mask + 1u);
    volatile uint32_t* vt = tab;

    uint32_t h = (key * 2654435761u) >> hash_shift;   // Fibonacci hash -> [0,cap)
    bool found = false;
    for (int probe = 0; probe < 8192; ++probe) {
        uint32_t cur = vt[h];                         // cheap pre-check
        if (cur == key) { found = true; break; }
        if (cur == 0u) {
            uint32_t prev = atomicCAS(&tab[h], 0u, key);
            if (prev == 0u) break;                    // we inserted it
            if (prev == key) { found = true; break; }
        }
        h = (h + 1u) & cap_mask;
    }
    if (!found) atomicAdd(&contrib[bin], 1u);         // first insert (or safety
                                                      // fallback on full table)
}

// ---------------------------------------------------------------------------
// Kernel 3: out = contrib > 0 ? acc / contrib : acc   (acc layout == out layout)
// ---------------------------------------------------------------------------
__global__ void finalize_kernel(const float* __restrict__ acc,
                                const uint32_t* __restrict__ contrib,
                                float* __restrict__ out, int n)
{
    int i = blockIdx.x * blockDim.x + threadIdx.x;
    if (i < n) {
        float a = acc[i];
        uint32_t c = contrib[i];
        out[i] = (c > 0u) ? (a / (float)c) : a;
    }
}

// ---------------------------------------------------------------------------
extern "C" void kernel_launch(void* const* d_in, const int* in_sizes, int n_in,
                              void* d_out, int out_size, void* d_ws, size_t ws_size,
                              hipStream_t stream)
{
    const float* flow   = (const float*)d_in[0];   // (B,2,H,W)
    const float* events = (const float*)d_in[1];   // (B,N,4)
    // d_in[2] (pol_mask) is exactly (p>0, p<0) of events[...,3]; derived on device.

    const int B         = in_sizes[0] / (2 * HW_DIM);
    const int nPerBatch = in_sizes[1] / (B * 4);
    const int nTotal    = B * nPerBatch;

    // Workspace layout: [acc: B*2*HW f32][contrib: B*2*HW u32][table: B*cap u32]
    const size_t nbins = (size_t)B * 2 * HW_DIM;
    float*    acc     = (float*)d_ws;
    uint32_t* contrib = (uint32_t*)d_ws + nbins;
    uint32_t* table   = contrib + nbins;

    const size_t used_bytes = nbins * 2 * sizeof(uint32_t);
    size_t avail_per_batch = 0;
    if (ws_size > used_bytes)
        avail_per_batch = (ws_size - used_bytes) / (sizeof(uint32_t) * (size_t)B);

    uint32_t cap = 1024u;                         // minimum (probe-limit fallback)
    while ((size_t)cap * 2 <= avail_per_batch && cap < (1u << 21)) cap <<= 1;
    const uint32_t cap_mask   = cap - 1u;
    const uint32_t hash_shift = 32u - (uint32_t)__builtin_ctz(cap);

    // 1) zero acc + contrib + table
    const size_t nwords = nbins * 2 + (size_t)B * cap;   // all multiples of 4
    const size_t nquads = nwords / 4;
    zero_ws_kernel<<<4096, 256, 0, stream>>>((uint4*)d_ws, nquads);

    // 2) event scatter (one 32-event tile per wave32)
    const int tiles  = (nTotal + 31) / 32;
    const int blocks = (tiles + WAVES_PER_BLOCK - 1) / WAVES_PER_BLOCK;
    iwe_scatter_kernel<<<blocks, WAVES_PER_BLOCK * 32, 0, stream>>>(
        flow, events, acc, contrib, table, nPerBatch, nTotal, cap_mask, hash_shift);

    // 3) finalize: averaged IWE into d_out (B,2,H,W)
    finalize_kernel<<<((int)nbins + 255) / 256, 256, 0, stream>>>(
        acc, contrib, (float*)d_out, (int)nbins);
}